// PaiIndexMatrixLSA_10934986736324
// MI455X (gfx1250) — compile-verified
//
#include <hip/hip_runtime.h>
#include <hip/hip_bf16.h>

typedef __attribute__((ext_vector_type(2))) float v2f;
typedef __attribute__((ext_vector_type(8))) float v8f;

#define NPTS   4096
#define NBATCH 8
#define KNN    20
#define NROWS  (NBATCH * NPTS)        // 32768
#define FEATD  140                    // 7*K
#define RFFD   128                    // 2*MAPPING
#define PERMN  400                    // 20*20

// ---------------------------------------------------------------------------
// Kernel 1: KNN top-20 per point. One batch's cloud (48KB) lives in LDS;
// each thread owns one query row and keeps a sorted top-20 in registers.
// Also emits spirals_index (= idx + b*N) as float into d_out[0:655360].
// ---------------------------------------------------------------------------
__global__ void __launch_bounds__(256) knn_kernel(const float* __restrict__ x,
                                                  int* __restrict__ idx_ws,
                                                  float* __restrict__ out_idx) {
    __shared__ float sx[NPTS], sy[NPTS], sz[NPTS];
    const int b      = blockIdx.x >> 4;          // 16 row-blocks per batch
    const int rowblk = blockIdx.x & 15;
    const float* xb = x + (size_t)b * 3 * NPTS;
    for (int i = threadIdx.x; i < NPTS; i += 256) {
        sx[i] = xb[i];
        sy[i] = xb[NPTS + i];
        sz[i] = xb[2 * NPTS + i];
    }
    __syncthreads();

    const int n = rowblk * 256 + threadIdx.x;    // query point within batch
    const float px = sx[n], py = sy[n], pz = sz[n];

    float best[KNN];
    int   bidx[KNN];
#pragma unroll
    for (int t = 0; t < KNN; ++t) { best[t] = -3.4e38f; bidx[t] = 0; }

    for (int m = 0; m < NPTS; ++m) {
        const float dx = px - sx[m];
        const float dy = py - sy[m];
        const float dz = pz - sz[m];
        const float d = -(dx * dx + dy * dy + dz * dz);   // neg squared dist
        if (d > best[KNN - 1]) {
            float cur = d; int ci = m;
#pragma unroll
            for (int t = 0; t < KNN; ++t) {               // strict > keeps
                const bool sw = cur > best[t];            // lower index on tie
                const float tf = best[t]; const int ti = bidx[t];
                if (sw) { best[t] = cur; bidx[t] = ci; cur = tf; ci = ti; }
            }
        }
    }

    const int bn   = b * NPTS + n;
    const int base = bn * KNN;
#pragma unroll
    for (int t = 0; t < KNN; ++t) {
        idx_ws[base + t]  = bidx[t];
        out_idx[base + t] = (float)(bidx[t] + b * NPTS);
    }
}

// ---------------------------------------------------------------------------
// Kernel 2: build x_feats (32768, 140): per (point, neighbor) -> 7 features
// ---------------------------------------------------------------------------
__global__ void __launch_bounds__(256) feats_kernel(const float* __restrict__ x,
                                                    const int* __restrict__ idx_ws,
                                                    float* __restrict__ feats) {
    const int g = blockIdx.x * 256 + threadIdx.x;     // (bn * 20 + k)
    const int bn = g / KNN;
    const int k  = g - bn * KNN;
    const int b  = bn >> 12;                          // /4096
    const float* xb = x + (size_t)b * 3 * NPTS;

    const int i0 = idx_ws[bn * KNN + 0];
    const int ik = idx_ws[bn * KNN + k];
    const float p0x = xb[i0], p0y = xb[NPTS + i0], p0z = xb[2 * NPTS + i0];
    const float qx  = xb[ik], qy  = xb[NPTS + ik], qz  = xb[2 * NPTS + ik];
    const float rx = qx - p0x, ry = qy - p0y, rz = qz - p0z;
    const float dis = sqrtf(rx * rx + ry * ry + rz * rz);

    float* f = feats + (size_t)bn * FEATD + k * 7;
    f[0] = p0x; f[1] = p0y; f[2] = p0z;
    f[3] = rx;  f[4] = ry;  f[5] = rz;  f[6] = dis;
}

// ---------------------------------------------------------------------------
// Kernel 3: WMMA GEMM proj = feats(32768,140) @ Bmat(140,64); rff = [sin,cos].
// One wave -> one 16x16 output tile. 2048 mtiles x 4 ntiles = 8192 waves.
// ---------------------------------------------------------------------------
__global__ void __launch_bounds__(256) rff_gemm_kernel(const float* __restrict__ feats,
                                                       const float* __restrict__ Bmat,
                                                       float* __restrict__ rff) {
    const int wave  = (blockIdx.x * 256 + threadIdx.x) >> 5;
    const int lane  = threadIdx.x & 31;
    const int ntile = wave & 3;
    const int mtile = wave >> 2;

    const int l16   = lane & 15;
    const int khalf = (lane >> 4) << 1;               // 0 or 2
    const int arow  = mtile * 16 + l16;               // A: M = lane%16
    const int col   = ntile * 16 + l16;               // B: N = lane%16

    v8f c = {};
    for (int kb = 0; kb < FEATD; kb += 4) {
        v2f a, bm;
        a.x  = feats[(size_t)arow * FEATD + kb + khalf];
        a.y  = feats[(size_t)arow * FEATD + kb + khalf + 1];
        bm.x = Bmat[(size_t)(kb + khalf) * 64 + col];
        bm.y = Bmat[(size_t)(kb + khalf + 1) * 64 + col];
        c = __builtin_amdgcn_wmma_f32_16x16x4_f32(false, a, false, bm,
                                                  (short)0, c, false, false);
    }

    const float TWO_PI = 6.28318530717958647692f;
    const int rbase = mtile * 16 + ((lane >> 4) ? 8 : 0);  // C/D row layout
#pragma unroll
    for (int r = 0; r < 8; ++r) {
        const float p = TWO_PI * c[r];
        const size_t rr = (size_t)(rbase + r) * RFFD;
        rff[rr + col]      = __sinf(p);
        rff[rr + 64 + col] = __cosf(p);
    }
}

// ---------------------------------------------------------------------------
// Kernel 4: logits = rff(32768,128) @ W^T(128,16) + b via WMMA, then
// sparsemax over 16 (one lane per row via LDS tile). 2048 waves.
// ---------------------------------------------------------------------------
__global__ void __launch_bounds__(256) attn_kernel(const float* __restrict__ rff,
                                                   const float* __restrict__ W,
                                                   const float* __restrict__ bias,
                                                   float* __restrict__ attn) {
    __shared__ float tile[8][16][17];                 // +1 pad vs bank conflicts
    const int wv    = threadIdx.x >> 5;
    const int lane  = threadIdx.x & 31;
    const int mtile = blockIdx.x * 8 + wv;

    const int l16   = lane & 15;
    const int khalf = (lane >> 4) << 1;
    const int arow  = mtile * 16 + l16;

    v8f c = {};
    for (int kb = 0; kb < RFFD; kb += 4) {
        v2f a, wb;
        a.x  = rff[(size_t)arow * RFFD + kb + khalf];
        a.y  = rff[(size_t)arow * RFFD + kb + khalf + 1];
        wb.x = W[(size_t)l16 * RFFD + kb + khalf];        // B[k][n] = W[n][k]
        wb.y = W[(size_t)l16 * RFFD + kb + khalf + 1];
        c = __builtin_amdgcn_wmma_f32_16x16x4_f32(false, a, false, wb,
                                                  (short)0, c, false, false);
    }

    const float bcol = bias[l16];
    const int rb = (lane >> 4) ? 8 : 0;
#pragma unroll
    for (int r = 0; r < 8; ++r) tile[wv][rb + r][l16] = c[r] + bcol;
    __syncthreads();

    if (lane < 16) {
        float z[16], zs[16];
#pragma unroll
        for (int i = 0; i < 16; ++i) { z[i] = tile[wv][lane][i]; zs[i] = z[i]; }
        // fully-unrolled bubble sort, descending
#pragma unroll
        for (int i = 0; i < 16; ++i)
#pragma unroll
            for (int j = 0; j < 15; ++j)
                if (zs[j] < zs[j + 1]) { float t = zs[j]; zs[j] = zs[j + 1]; zs[j + 1] = t; }
        // sparsemax threshold
        float cs = 0.0f, csk = 0.0f; int kk = 1;
#pragma unroll
        for (int j = 0; j < 16; ++j) {
            cs += zs[j];
            if (1.0f + (float)(j + 1) * zs[j] > cs) { kk = j + 1; csk = cs; }
        }
        const float tau = (csk - 1.0f) / (float)kk;
        float* ap = attn + (size_t)(mtile * 16 + lane) * 16;
#pragma unroll
        for (int i = 0; i < 16; ++i) ap[i] = fmaxf(z[i] - tau, 0.0f);
    }
}

// ---------------------------------------------------------------------------
// Kernel 5: permatrix = attn(32768,16) @ perm(16,400) via WMMA.
// 2048 mtiles x 25 ntiles = 51200 waves. K=16 -> 4 WMMA steps.
// ---------------------------------------------------------------------------
__global__ void __launch_bounds__(256) perm_gemm_kernel(const float* __restrict__ attn,
                                                        const float* __restrict__ perm,
                                                        float* __restrict__ out) {
    const int wave  = (blockIdx.x * 256 + threadIdx.x) >> 5;
    const int mtile = wave / 25;
    const int ntile = wave - mtile * 25;

    const int lane  = threadIdx.x & 31;
    const int l16   = lane & 15;
    const int khalf = (lane >> 4) << 1;
    const int arow  = mtile * 16 + l16;
    const int col   = ntile * 16 + l16;

    v8f c = {};
#pragma unroll
    for (int kb = 0; kb < 16; kb += 4) {
        v2f a, pm;
        a.x  = attn[(size_t)arow * 16 + kb + khalf];
        a.y  = attn[(size_t)arow * 16 + kb + khalf + 1];
        pm.x = perm[(size_t)(kb + khalf) * PERMN + col];
        pm.y = perm[(size_t)(kb + khalf + 1) * PERMN + col];
        c = __builtin_amdgcn_wmma_f32_16x16x4_f32(false, a, false, pm,
                                                  (short)0, c, false, false);
    }

    const int rbase = mtile * 16 + ((lane >> 4) ? 8 : 0);
#pragma unroll
    for (int r = 0; r < 8; ++r)
        out[(size_t)(rbase + r) * PERMN + col] = c[r];
}

// ---------------------------------------------------------------------------
extern "C" void kernel_launch(void* const* d_in, const int* in_sizes, int n_in,
                              void* d_out, int out_size, void* d_ws, size_t ws_size,
                              hipStream_t stream) {
    const float* x    = (const float*)d_in[0];   // (8,3,4096)
    const float* Bmat = (const float*)d_in[1];   // (140,64)
    const float* W    = (const float*)d_in[2];   // (16,128)
    const float* bias = (const float*)d_in[3];   // (16,)
    const float* perm = (const float*)d_in[4];   // (16,20,20) -> (16,400)

    float* out_idx  = (float*)d_out;                          // 655360 floats
    float* out_perm = (float*)d_out + (size_t)NROWS * KNN;    // 32768*400

    char* ws = (char*)d_ws;
    int*   idx_ws = (int*)ws;                                          // 2.62 MB
    float* feats  = (float*)(ws + (size_t)NROWS * KNN * 4);            // 18.35 MB
    float* rff    = (float*)(ws + (size_t)NROWS * KNN * 4
                                + (size_t)NROWS * FEATD * 4);          // 16.78 MB
    float* attn   = (float*)(ws + (size_t)NROWS * KNN * 4
                                + (size_t)NROWS * FEATD * 4
                                + (size_t)NROWS * RFFD * 4);           // 2.10 MB

    // 1) KNN: 8 batches x 16 row-blocks
    knn_kernel<<<dim3(NBATCH * 16), dim3(256), 0, stream>>>(x, idx_ws, out_idx);

    // 2) features: 32768*20 threads
    feats_kernel<<<dim3((NROWS * KNN) / 256), dim3(256), 0, stream>>>(x, idx_ws, feats);

    // 3) RFF GEMM: 2048 mtiles * 4 ntiles = 8192 waves -> 1024 blocks
    rff_gemm_kernel<<<dim3((NROWS / 16) * 4 * 32 / 256), dim3(256), 0, stream>>>(feats, Bmat, rff);

    // 4) attention logits + sparsemax: 2048 waves -> 256 blocks
    attn_kernel<<<dim3(NROWS / 16 / 8), dim3(256), 0, stream>>>(rff, W, bias, attn);

    // 5) permatrix GEMM: 2048 * 25 waves -> 6400 blocks
    perm_gemm_kernel<<<dim3((NROWS / 16) * 25 * 32 / 256), dim3(256), 0, stream>>>(attn, perm, out_perm);
}